// SampleFeat_26079041421815
// MI455X (gfx1250) — compile-verified
//
#include <hip/hip_runtime.h>
#include <cstdint>
#include <cstddef>

// Problem constants (from reference setup_inputs).
constexpr int Bn  = 4;
constexpr int Cn  = 256;
constexpr int Hn  = 100;
constexpr int Wn  = 100;
constexpr int Pn  = 9;
constexpr int HWn = Hn * Wn;

// One thread owns one (b, p, pixel) and loops over all 256 channels.
// Weights/indices are computed once per pixel and reused across channels.
__global__ __launch_bounds__(256, 4) void deform_sample_kernel(
    const float* __restrict__ x,       // (B, C, H, W)
    const float* __restrict__ offset,  // (B, P*2, H, W)  [p*2+0]=fy, [p*2+1]=fx
    float* __restrict__ out)           // (B, P*C, H, W)
{
    const int p   = blockIdx.y;
    const int b   = blockIdx.z;
    const int tid = threadIdx.x;

    int pix = blockIdx.x * blockDim.x + tid;
    const bool active = (pix < HWn);
    if (pix >= HWn) pix = HWn - 1;   // clamp (no divergence before asm)

    // ---- CDNA5 async global->LDS staging of this thread's two offsets ----
    __shared__ float s_off[2 * 256];
    {
        const float* gy = offset + ((size_t)(b * Pn + p) * 2 + 0) * HWn + pix;
        const float* gx = gy + HWn;
        unsigned lds_y = (unsigned)(unsigned long long)(&s_off[tid]);
        unsigned lds_x = (unsigned)(unsigned long long)(&s_off[256 + tid]);
        unsigned long long ga_y = (unsigned long long)gy;
        unsigned long long ga_x = (unsigned long long)gx;
        asm volatile("global_load_async_to_lds_b32 %0, %1, off"
                     :: "v"(lds_y), "v"(ga_y) : "memory");
        asm volatile("global_load_async_to_lds_b32 %0, %1, off"
                     :: "v"(lds_x), "v"(ga_x) : "memory");
        asm volatile("s_wait_asynccnt 0" ::: "memory");
    }
    const float fy = s_off[tid];
    const float fx = s_off[256 + tid];

    // ---- per-pixel bilinear setup (matches reference semantics) ----
    const int h = pix / Wn;
    const int w = pix - h * Wn;

    const float px  = fx + (float)w;
    const float py  = fy + (float)h;
    const float x0f = floorf(px);
    const float y0f = floorf(py);
    const float wx1 = px - x0f;
    const float wy1 = py - y0f;
    const float wx0 = 1.0f - wx1;
    const float wy0 = 1.0f - wy1;

    const int x0 = (int)x0f;
    const int y0 = (int)y0f;
    const int x1 = x0 + 1;
    const int y1 = y0 + 1;

    const float vx0 = (x0 >= 0 && x0 < Wn) ? 1.0f : 0.0f;
    const float vx1 = (x1 >= 0 && x1 < Wn) ? 1.0f : 0.0f;
    const float vy0 = (y0 >= 0 && y0 < Hn) ? 1.0f : 0.0f;
    const float vy1 = (y1 >= 0 && y1 < Hn) ? 1.0f : 0.0f;

    const int cx0 = min(max(x0, 0), Wn - 1);
    const int cx1 = min(max(x1, 0), Wn - 1);
    const int cy0 = min(max(y0, 0), Hn - 1);
    const int cy1 = min(max(y1, 0), Hn - 1);

    const int i00 = cy0 * Wn + cx0;
    const int i01 = cy0 * Wn + cx1;
    const int i10 = cy1 * Wn + cx0;
    const int i11 = cy1 * Wn + cx1;

    const float w00 = wx0 * wy0 * (vx0 * vy0);
    const float w01 = wx1 * wy0 * (vx1 * vy0);
    const float w10 = wx0 * wy1 * (vx0 * vy1);
    const float w11 = wx1 * wy1 * (vx1 * vy1);

    if (!active) return;

    const float* xb = x + (size_t)b * Cn * HWn;
    const float* p00 = xb + i00;
    const float* p01 = xb + i01;
    const float* p10 = xb + i10;
    const float* p11 = xb + i11;
    float* ob = out + ((size_t)(b * Pn + p) * Cn) * HWn + pix;

    // Channel loop: 4 gathers + 1 NT store per channel. x stays L2-resident
    // (41 MB << 192 MB L2); NT stores keep the 368 MB output from evicting it.
#pragma unroll 8
    for (int c = 0; c < Cn; ++c) {
        const size_t co = (size_t)c * HWn;
        // speculative prefetch 16 channels ahead (tail over-reach is dropped)
        __builtin_prefetch(p00 + co + (size_t)16 * HWn, 0, 0);
        float v = p00[co] * w00 + p01[co] * w01 +
                  p10[co] * w10 + p11[co] * w11;
        __builtin_nontemporal_store(v, ob + co);
    }
}

extern "C" void kernel_launch(void* const* d_in, const int* in_sizes, int n_in,
                              void* d_out, int out_size, void* d_ws, size_t ws_size,
                              hipStream_t stream) {
    (void)in_sizes; (void)n_in; (void)out_size; (void)d_ws; (void)ws_size;
    const float* x      = (const float*)d_in[0];
    const float* offset = (const float*)d_in[1];
    float* out          = (float*)d_out;

    dim3 block(256, 1, 1);
    dim3 grid((HWn + 255) / 256, Pn, Bn);   // (40, 9, 4)
    deform_sample_kernel<<<grid, block, 0, stream>>>(x, offset, out);
}